// LossConsistenciaMorfologicaCompuesta_91010357002868
// MI455X (gfx1250) — compile-verified
//
#include <hip/hip_runtime.h>
#include <hip/hip_bf16.h>

#define B_      16
#define H_      1024
#define W_      1024
#define TILE    64
#define HALO    3
#define LW      (TILE + 2 * HALO)   // 70
#define LP      72                  // padded LDS row stride (words)
#define NTHREADS 256
#define NACC    13
#define NBLOCKS (B_ * (H_ / TILE) * (W_ / TILE)) // 4096

#define _AS1 __attribute__((address_space(1)))
#define _AS3 __attribute__((address_space(3)))

#if defined(__HIP_DEVICE_COMPILE__) && defined(__gfx1250__) && \
    __has_builtin(__builtin_amdgcn_global_load_async_to_lds_b32)
#define USE_ASYNC_LDS 1
#else
#define USE_ASYNC_LDS 0
#endif

__device__ __forceinline__ float wave_sum(float v) {
#pragma unroll
  for (int off = 16; off > 0; off >>= 1)
    v += __shfl_down(v, off, 32);
  return v;
}

__global__ __launch_bounds__(NTHREADS)
void morph_dice_partial(const float* __restrict__ pred,
                        const float* __restrict__ teacher,
                        float* __restrict__ partials) {
  __shared__ float sm[LW * LP];

  const int bx = blockIdx.x;   // tile col 0..15
  const int by = blockIdx.y;   // tile row 0..15
  const int bz = blockIdx.z;   // batch    0..15
  const int tid = threadIdx.x;

  const size_t planeBase = (size_t)bz * (size_t)(H_ * W_);
  const int gx0 = bx * TILE - HALO;
  const int gy0 = by * TILE - HALO;

  // ---- Stage teacher halo tile into LDS (clamp-to-edge == geodesic pad) ----
  for (int idx = tid; idx < LW * LW; idx += NTHREADS) {
    const int r = idx / LW;
    const int c = idx - r * LW;
    int gy = gy0 + r; gy = gy < 0 ? 0 : (gy > H_ - 1 ? H_ - 1 : gy);
    int gx = gx0 + c; gx = gx < 0 ? 0 : (gx > W_ - 1 ? W_ - 1 : gx);
    const float* gptr = teacher + planeBase + (size_t)gy * W_ + gx;
    float* lptr = &sm[r * LP + c];
#if USE_ASYNC_LDS
    float* gmut = const_cast<float*>(gptr);
    __builtin_amdgcn_global_load_async_to_lds_b32(
        (_AS1 int*)gmut, (_AS3 int*)lptr, 0, 0);
#else
    *lptr = *gptr;
#endif
  }
#if USE_ASYNC_LDS
  asm volatile("s_wait_asynccnt 0" ::: "memory");
#endif
  __syncthreads();

  // ---- Compute: each thread owns one column, 16 rows ----
  const int c  = tid & (TILE - 1);  // 0..63
  const int rg = tid >> 6;          // 0..3
  const int r0 = rg * 16;           // local row base
  const int lx = c + HALO;          // LDS x index

  float h1[7], h3x[7], h3n[7], h5x[7], h5n[7];

  auto loadrow = [&](int lr, int slot) {
    const float* row = &sm[lr * LP + lx];
    const float xm2 = row[-2], xm1 = row[-1], x0 = row[0];
    const float xp1 = row[1],  xp2 = row[2];
    h1[slot] = x0;
    const float px = fmaxf(xm1, xp1), pn = fminf(xm1, xp1);
    const float t3x = fmaxf(px, x0),  t3n = fminf(pn, x0);
    h3x[slot] = t3x; h3n[slot] = t3n;
    const float qx = fmaxf(xm2, xp2), qn = fminf(xm2, xp2);
    h5x[slot] = fmaxf(t3x, qx);
    h5n[slot] = fminf(t3n, qn);
  };

  // Preload LDS rows r0..r0+5 into window slots 1..6
#pragma unroll
  for (int i = 0; i < 6; ++i) loadrow(r0 + i, i + 1);

  float sp = 0.f;
  float st[6]  = {0.f, 0.f, 0.f, 0.f, 0.f, 0.f};
  float spt[6] = {0.f, 0.f, 0.f, 0.f, 0.f, 0.f};

  const float* predCol =
      pred + planeBase + (size_t)(by * TILE + r0) * W_ + (bx * TILE + c);

#pragma unroll
  for (int s = 0; s < 16; ++s) {
    // shift rolling window
#pragma unroll
    for (int i = 0; i < 6; ++i) {
      h1[i] = h1[i + 1];  h3x[i] = h3x[i + 1]; h3n[i] = h3n[i + 1];
      h5x[i] = h5x[i + 1]; h5n[i] = h5n[i + 1];
    }
    loadrow(r0 + 6 + s, 6);  // new bottom row

    // width-7 horizontal extent only needed at the center row
    const int lrC = r0 + s + 3;
    const float xm3 = sm[lrC * LP + lx - 3];
    const float xp3 = sm[lrC * LP + lx + 3];
    const float h7x = fmaxf(h5x[3], fmaxf(xm3, xp3));
    const float h7n = fminf(h5n[3], fminf(xm3, xp3));

    // vertical combine per structuring element (row widths: see analysis)
    const float m3x = fmaxf(h3x[3], fmaxf(h1[2], h1[4]));
    const float m3n = fminf(h3n[3], fminf(h1[2], h1[4]));
    const float m5x = fmaxf(fmaxf(h5x[3], fmaxf(h3x[2], h3x[4])),
                            fmaxf(h1[1], h1[5]));
    const float m5n = fminf(fminf(h5n[3], fminf(h3n[2], h3n[4])),
                            fminf(h1[1], h1[5]));
    const float m7x = fmaxf(fmaxf(h7x, fmaxf(h5x[2], h5x[4])),
                            fmaxf(fmaxf(h5x[1], h5x[5]), fmaxf(h1[0], h1[6])));
    const float m7n = fminf(fminf(h7n, fminf(h5n[2], h5n[4])),
                            fminf(fminf(h5n[1], h5n[5]), fminf(h1[0], h1[6])));

    const float p = predCol[(size_t)s * W_];
    sp += p;
    st[0] += m3x; spt[0] = fmaf(p, m3x, spt[0]);
    st[1] += m3n; spt[1] = fmaf(p, m3n, spt[1]);
    st[2] += m5x; spt[2] = fmaf(p, m5x, spt[2]);
    st[3] += m5n; spt[3] = fmaf(p, m5n, spt[3]);
    st[4] += m7x; spt[4] = fmaf(p, m7x, spt[4]);
    st[5] += m7n; spt[5] = fmaf(p, m7n, spt[5]);
  }

  // ---- Block reduction: wave32 shuffle -> LDS -> 13 floats per block ----
  float vals[NACC];
  vals[0] = sp;
#pragma unroll
  for (int i = 0; i < 6; ++i) { vals[1 + 2 * i] = st[i]; vals[2 + 2 * i] = spt[i]; }

  __syncthreads();  // done reading sm; safe to reuse
  const int lane = tid & 31;
  const int wid  = tid >> 5;  // 0..7
#pragma unroll
  for (int i = 0; i < NACC; ++i) {
    const float w = wave_sum(vals[i]);
    if (lane == 0) sm[wid * NACC + i] = w;
  }
  __syncthreads();
  if (tid < NACC) {
    float s = 0.f;
#pragma unroll
    for (int w = 0; w < 8; ++w) s += sm[w * NACC + tid];
    const int blk = (bz * gridDim.y + by) * gridDim.x + bx;
    partials[blk * NACC + tid] = s;
  }
}

__global__ __launch_bounds__(256)
void reduce_finalize(const float* __restrict__ partials, float* __restrict__ out) {
  __shared__ double red[256];
  const int tid = threadIdx.x;

  double tot[NACC];
  for (int j = 0; j < NACC; ++j) {
    double s = 0.0;
    for (int b = tid; b < NBLOCKS; b += 256)
      s += (double)partials[b * NACC + j];
    red[tid] = s;
    __syncthreads();
    for (int off = 128; off > 0; off >>= 1) {
      if (tid < off) red[tid] += red[tid + off];
      __syncthreads();
    }
    if (tid == 0) tot[j] = red[0];
    __syncthreads();
  }

  if (tid == 0) {
    const double sp = tot[0];
    double loss = 0.0;
    for (int i = 0; i < 6; ++i) {
      const double stv  = tot[1 + 2 * i];
      const double sptv = tot[2 + 2 * i];
      double card = sp + stv;
      card = card > 1e-7 ? card : 1e-7;
      const double score = 2.0 * sptv / card;
      loss += (stv > 0.0) ? (1.0 - score) : 0.0;
    }
    out[0] = (float)(loss / 3.0);
  }
}

extern "C" void kernel_launch(void* const* d_in, const int* in_sizes, int n_in,
                              void* d_out, int out_size, void* d_ws, size_t ws_size,
                              hipStream_t stream) {
  (void)in_sizes; (void)n_in; (void)out_size; (void)ws_size;
  const float* pred    = (const float*)d_in[0];   // pred_student_prob
  const float* teacher = (const float*)d_in[1];   // teacher_prob
  float* partials = (float*)d_ws;                 // NBLOCKS * NACC floats

  dim3 grid(W_ / TILE, H_ / TILE, B_);
  morph_dice_partial<<<grid, dim3(NTHREADS), 0, stream>>>(pred, teacher, partials);
  reduce_finalize<<<dim3(1), dim3(256), 0, stream>>>(partials, (float*)d_out);
}